// DLRM_interact_features_70781061038447
// MI455X (gfx1250) — compile-verified
//
#include <hip/hip_runtime.h>
#include <hip/hip_bf16.h>

// ext-vector types (HIP's float4 is a struct; we need real vectors for
// __builtin_nontemporal_store and the WMMA builtin operands)
typedef float v8f __attribute__((ext_vector_type(8)));
typedef float v2f __attribute__((ext_vector_type(2)));
typedef float f4  __attribute__((ext_vector_type(4)));

#define GRIDN  4096
#define BLOCKN 256   // 8 wave32s

// Kernel A: fused streaming copy (x -> out) + f32 WMMA Gram accumulation.
// Virtual row m = 4*seg + vec (16 rows = 4 K-segments x 4 vectors).
// One v_wmma_f32_16x16x4_f32 consumes 4 K-elements from 4 segments of all
// 4 vectors (64 floats / 256 B per wave). Gram is symmetric and the ISA's
// A-layout (VGPR0=K0|K2, VGPR1=K1|K3 across lane halves) transposes onto the
// B-layout, so the same register pair is passed as both A and B.
__global__ void __launch_bounds__(BLOCKN)
dlrm_gram_copy(const float* __restrict__ x,  const float* __restrict__ l0,
               const float* __restrict__ l1, const float* __restrict__ l2,
               float* __restrict__ out, float* __restrict__ blockZ,
               int tileLen, int itersPerWave)
{
    __shared__ float cb[8][32][8];   // [wave][lane][accum-vgpr]  (8 KB LDS)

    const int tid  = threadIdx.x;
    const int lane = tid & 31;       // wave32
    const int wave = tid >> 5;
    const long long tileStart = (long long)blockIdx.x * (long long)tileLen;

    // ---- streaming copy of this tile of x -> out ----
    // nontemporal store: out is write-once, keep L2 capacity for the inputs
    {
        const f4* __restrict__ src = reinterpret_cast<const f4*>(x + tileStart);
        f4* dst = reinterpret_cast<f4*>(out + tileStart);
        const int n4 = tileLen >> 2;
        for (int i = tid; i < n4; i += BLOCKN) {
            f4 v = src[i];
            __builtin_nontemporal_store(v, dst + i);
        }
    }

    // ---- WMMA f32 Gram over this tile ----
    const int r   = lane & 15;
    const int vec = r & 3;           // which of the 4 vectors
    const int seg = r >> 2;          // which K-segment (0..3)
    const float* vp = (vec == 0) ? x : (vec == 1) ? l0 : (vec == 2) ? l1 : l2;

    v8f c = {};                                  // 16x16 f32 accumulator
    long long base = tileStart + (long long)(wave * 16 + seg * 4);
    for (int it = 0; it < itersPerWave; ++it) {  // uniform trip count: EXEC all-1s
        f4 e = *reinterpret_cast<const f4*>(vp + base);   // global_load_b128
        v2f a;
        a.x = (lane < 16) ? e.x : e.z;           // A VGPR0: K=0 | K=2
        a.y = (lane < 16) ? e.y : e.w;           // A VGPR1: K=1 | K=3
        // D = A*A(sym) + C ; v_wmma_f32_16x16x4_f32
        c = __builtin_amdgcn_wmma_f32_16x16x4_f32(
                /*neg_a=*/false, a, /*neg_b=*/false, a,
                /*c_mod=*/(short)0, c, /*reuse_a=*/false, /*reuse_b=*/false);
        base += 8 * 16;                          // 8 waves x 16 elems / iter
    }

    // spill accumulator to LDS, then fixed-order (deterministic) extraction of
    // the four diagonal 4x4 blocks: Z[i][j] = sum_s C[4s+i][4s+j]
    *reinterpret_cast<v8f*>(&cb[wave][lane][0]) = c;
    __syncthreads();

    if (tid < 16) {
        const int i = tid >> 2, j = tid & 3;     // entry = 4*i + j
        float z = 0.0f;
        #pragma unroll
        for (int w = 0; w < 8; ++w) {
            #pragma unroll
            for (int s = 0; s < 4; ++s) {
                const int m = 4 * s + i, n = 4 * s + j;
                // C/D layout: VGPR g -> M=g (lanes 0-15) or M=g+8 (lanes 16-31)
                const int g  = (m < 8) ? m : (m - 8);
                const int ln = (m < 8) ? n : (n + 16);
                z += cb[w][ln][g];
            }
        }
        blockZ[(long long)blockIdx.x * 16 + tid] = z;
    }
}

// Kernel B: deterministic fixed-order reduction of per-block Gram partials,
// then emit the 6 strict-lower-triangular entries (np.tril_indices(4, -1)).
__global__ void __launch_bounds__(BLOCKN)
dlrm_reduce(const float* __restrict__ blockZ, float* __restrict__ out,
            int nBlocks, long long D)
{
    __shared__ float acc[16][17];
    const int tid = threadIdx.x;
    const int e = tid >> 4, u = tid & 15;        // 16 threads per Z entry
    float s = 0.0f;
    for (int b = u; b < nBlocks; b += 16)
        s += blockZ[(long long)b * 16 + e];
    acc[e][u] = s;
    __syncthreads();
    if (tid < 16) {
        float t = 0.0f;
        #pragma unroll
        for (int k = 0; k < 16; ++k) t += acc[tid][k];
        acc[tid][16] = t;
    }
    __syncthreads();
    if (tid < 6) {
        const int li[6] = {1, 2, 2, 3, 3, 3};
        const int lj[6] = {0, 0, 1, 0, 1, 2};
        out[D + tid] = acc[li[tid] * 4 + lj[tid]][16];
    }
}

extern "C" void kernel_launch(void* const* d_in, const int* in_sizes, int n_in,
                              void* d_out, int out_size, void* d_ws, size_t ws_size,
                              hipStream_t stream) {
    const float* x  = (const float*)d_in[0];
    const float* l0 = (const float*)d_in[1];
    const float* l1 = (const float*)d_in[2];
    const float* l2 = (const float*)d_in[3];
    float* out = (float*)d_out;
    float* blockZ = (float*)d_ws;                // GRIDN * 16 floats = 256 KB

    const long long D = (long long)in_sizes[0];  // 16777216 (2^24)
    const int tileLen = (int)(D / GRIDN);        // 4096
    const int itersPerWave = tileLen / (8 * 16); // 32

    dlrm_gram_copy<<<GRIDN, BLOCKN, 0, stream>>>(x, l0, l1, l2, out, blockZ,
                                                 tileLen, itersPerWave);
    dlrm_reduce<<<1, BLOCKN, 0, stream>>>(blockZ, out, GRIDN, D);
}